// LrmcSeededPoolGCN_21964462752257
// MI455X (gfx1250) — compile-verified
//
#include <hip/hip_runtime.h>

#define IN_F 128
#define H_F  64
#define OUT_F 64
#define KC   1024

typedef float v2f __attribute__((ext_vector_type(2)));
typedef float v8f __attribute__((ext_vector_type(8)));

// ---- fp32 WMMA row-strip: acc[4] (16x64) += A(16xKD) @ B(KDx64) ----
// A fragment loaded once per K-step, reused across the 4 column tiles.
// INTERLEAVED: B element (k,col) lives at ((k>>1)*64+col)*2+(k&1)  (LDS path)
// so each B fragment is one contiguous 8-byte ds_load_b64 (no repack movs).
template <int KD, bool INTERLEAVED>
__device__ __forceinline__ void rs_mac(const float* __restrict__ arow,
                                       const float* __restrict__ B,
                                       int lr, int kh, v8f* acc) {
  for (int k = 0; k < KD; k += 4) {
    v2f a;
    a.x = arow[k + 0];
    a.y = arow[k + 1];
#pragma unroll
    for (int t = 0; t < 4; ++t) {
      const int col = t * 16 + lr;
      v2f b;
      if constexpr (INTERLEAVED) {
        b = *(const v2f*)(B + (size_t)((k + kh) >> 1) * 128 + col * 2);
      } else {
        b.x = B[(size_t)(k + kh + 0) * 64 + col];
        b.y = B[(size_t)(k + kh + 1) * 64 + col];
      }
      acc[t] = __builtin_amdgcn_wmma_f32_16x16x4_f32(false, a, false, b,
                                                     (short)0, acc[t], false, false);
    }
  }
}

// ---- unified row-strip GEMM: C(Mx64) = A(MxKD) @ B(KDx64) (+bias) ----
template <int KD, bool BIAS>
__global__ void __launch_bounds__(256) gemm_rs(const float* __restrict__ A, int lda,
                                               const float* __restrict__ Bg,
                                               const float* __restrict__ bias,
                                               float* __restrict__ C, int Mrows) {
  constexpr bool USE_LDS = (KD <= 128);           // B <= 32 KB -> stage in LDS
  __shared__ float ldsB[USE_LDS ? KD * 64 : 1];
  if constexpr (USE_LDS) {
    for (int j = threadIdx.x; j < KD * 64; j += 256) {
      const int k = j >> 6, col = j & 63;         // coalesced read, permuted write
      ldsB[(((k >> 1) * 64 + col) << 1) + (k & 1)] = Bg[j];
    }
    __syncthreads();
  }
  const int wave = threadIdx.x >> 5, lane = threadIdx.x & 31;
  const int tile = blockIdx.x * 8 + wave;
  if (tile >= (Mrows >> 4)) return;               // wave-uniform: EXEC all-1s
  const int row0 = tile << 4;
  const int lr = lane & 15, kh = (lane >> 4) << 1, half = lane >> 4;
  v8f acc[4] = {};
  const float* arow = A + (size_t)(row0 + lr) * lda + kh;
  if constexpr (USE_LDS) rs_mac<KD, true >(arow, ldsB, lr, kh, acc);
  else                   rs_mac<KD, false>(arow, Bg,   lr, kh, acc);
#pragma unroll
  for (int t = 0; t < 4; ++t) {
    const int col = t * 16 + lr;
    const float bb = BIAS ? bias[col] : 0.0f;
#pragma unroll
    for (int r = 0; r < 8; ++r)
      C[(size_t)(row0 + r + 8 * half) * 64 + col] = acc[t][r] + bb;
  }
}

// ---- fused: out = xp2[cid[n]] + relu(alpha) * (x1 @ Wskip + bskip) ----
__global__ void __launch_bounds__(256) skip_out(const float* __restrict__ X1,
                                                const float* __restrict__ Wsk,
                                                const float* __restrict__ bsk,
                                                const float* __restrict__ xp2,
                                                const int* __restrict__ cid,
                                                const float* __restrict__ alpha,
                                                float* __restrict__ outp, int n) {
  __shared__ float ldsB[H_F * 64];                // Wskip: 16 KB, interleaved
  for (int j = threadIdx.x; j < H_F * 64; j += 256) {
    const int k = j >> 6, col = j & 63;
    ldsB[(((k >> 1) * 64 + col) << 1) + (k & 1)] = Wsk[j];
  }
  __syncthreads();
  const int wave = threadIdx.x >> 5, lane = threadIdx.x & 31;
  const int tile = blockIdx.x * 8 + wave;
  if (tile >= (n >> 4)) return;
  const int row0 = tile << 4;
  const int lr = lane & 15, kh = (lane >> 4) << 1, half = lane >> 4;
  v8f acc[4] = {};
  const float* arow = X1 + (size_t)(row0 + lr) * H_F + kh;
  rs_mac<H_F, true>(arow, ldsB, lr, kh, acc);
  const float a = fmaxf(alpha[0], 0.0f);
#pragma unroll
  for (int r = 0; r < 8; ++r) {
    const int row = row0 + r + 8 * half;
    const int c = cid[row];
#pragma unroll
    for (int t = 0; t < 4; ++t) {
      const int col = t * 16 + lr;
      outp[(size_t)row * OUT_F + col] =
          xp2[(size_t)c * OUT_F + col] + a * (acc[t][r] + bsk[col]);
    }
  }
}

// ---- irregular / elementwise kernels ----
__global__ void deg_count(const int* __restrict__ dst, float* __restrict__ deg, int E) {
  int e = blockIdx.x * blockDim.x + threadIdx.x;
  if (e < E) atomicAdd(&deg[dst[e]], 1.0f);
}

__global__ void deg_to_dinv(float* __restrict__ deg, int n) {
  int i = blockIdx.x * blockDim.x + threadIdx.x;
  if (i < n) deg[i] = rsqrtf(deg[i] + 1.0f);   // deg becomes dinv in place
}

__global__ void edge_agg(const int* __restrict__ src, const int* __restrict__ dst,
                         const float* __restrict__ dinv, const float* __restrict__ Hh,
                         float* __restrict__ agg, int E) {
  long long tid = (long long)blockIdx.x * blockDim.x + threadIdx.x;
  int e = (int)(tid >> 4), g = (int)(tid & 15);   // 16 lanes x float4 = 64 feats
  if (e >= E) return;
  int s = src[e], d = dst[e];
  float norm = dinv[s] * dinv[d];
  float4 hv = *(const float4*)(Hh + (size_t)s * H_F + g * 4);
  float* ap = agg + (size_t)d * H_F + g * 4;
  atomicAdd(ap + 0, hv.x * norm);
  atomicAdd(ap + 1, hv.y * norm);
  atomicAdd(ap + 2, hv.z * norm);
  atomicAdd(ap + 3, hv.w * norm);
}

__global__ void x1_pool(const float* __restrict__ agg, const float* __restrict__ Hh,
                        const float* __restrict__ dinv, const float* __restrict__ b1,
                        const int* __restrict__ cid, float* __restrict__ x1,
                        float* __restrict__ sums, float* __restrict__ counts, int n) {
  long long tid = (long long)blockIdx.x * blockDim.x + threadIdx.x;
  int i = (int)(tid >> 6), f = (int)(tid & 63);
  if (i >= n) return;
  float di = dinv[i];
  float v = agg[(size_t)i * H_F + f] + di * di * Hh[(size_t)i * H_F + f] + b1[f];
  v = fmaxf(v, 0.0f);
  x1[(size_t)i * H_F + f] = v;
  int c = cid[i];
  atomicAdd(&sums[(size_t)c * H_F + f], v);
  if (f == 0) atomicAdd(&counts[c], 1.0f);
}

__global__ void xp_mean(const float* __restrict__ sums, const float* __restrict__ counts,
                        float* __restrict__ xp) {
  int t = blockIdx.x * blockDim.x + threadIdx.x;
  if (t < KC * H_F) xp[t] = sums[t] / fmaxf(counts[t >> 6], 1.0f);
}

__global__ void a_build(const int* __restrict__ src, const int* __restrict__ dst,
                        const int* __restrict__ cid, float* __restrict__ A, int E) {
  int e = blockIdx.x * blockDim.x + threadIdx.x;
  if (e >= E) return;
  int cu = cid[src[e]], cv = cid[dst[e]];
  if (cu != cv) A[(size_t)cu * KC + cv] = 1.0f;   // idempotent racing stores
}

__global__ void dinvp_k(const float* __restrict__ A, float* __restrict__ dinvp) {
  int t = blockIdx.x * blockDim.x + threadIdx.x;
  if (t >= KC) return;
  float s = 1.0f;                                  // self-loop on the diagonal
  for (int r = 0; r < KC; ++r) s += A[(size_t)r * KC + t];  // coalesced over t
  dinvp[t] = rsqrtf(s);
}

__global__ void an_build(const float* __restrict__ A, const float* __restrict__ dinvp,
                         float* __restrict__ An) {
  int tid = blockIdx.x * blockDim.x + threadIdx.x;
  if (tid >= KC * KC) return;
  int t = tid >> 10, s = tid & (KC - 1);
  float v = A[(size_t)s * KC + t] + (s == t ? 1.0f : 0.0f);
  An[tid] = v * dinvp[s] * dinvp[t];               // An[t][s] = A_norm[s][t]
}

extern "C" void kernel_launch(void* const* d_in, const int* in_sizes, int n_in,
                              void* d_out, int out_size, void* d_ws, size_t ws_size,
                              hipStream_t stream) {
  const float* x     = (const float*)d_in[0];
  const int*   edge  = (const int*)  d_in[1];
  const int*   cid   = (const int*)  d_in[2];
  const float* W1    = (const float*)d_in[3];
  const float* b1    = (const float*)d_in[4];
  const float* W2    = (const float*)d_in[5];
  const float* b2    = (const float*)d_in[6];
  const float* Wsk   = (const float*)d_in[7];
  const float* bsk   = (const float*)d_in[8];
  const float* alpha = (const float*)d_in[9];
  const int n = in_sizes[0] / IN_F;
  const int E = in_sizes[1] / 2;
  const int* srcp = edge;
  const int* dstp = edge + E;

  float* ws = (float*)d_ws;
  size_t off = 0;
  float* Hh    = ws + off; off += (size_t)n * H_F;
  float* x1    = ws + off; off += (size_t)n * H_F;
  float* xp    = ws + off; off += (size_t)KC * H_F;
  float* An    = ws + off; off += (size_t)KC * KC;
  float* dinvp = ws + off; off += KC;
  float* hp    = ws + off; off += (size_t)KC * H_F;
  float* xp2   = ws + off; off += (size_t)KC * H_F;
  // contiguous zero-initialized region:
  float* zbase  = ws + off;
  float* agg    = ws + off; off += (size_t)n * H_F;
  float* deg    = ws + off; off += (size_t)n;          // becomes dinv
  float* sums   = ws + off; off += (size_t)KC * H_F;
  float* counts = ws + off; off += (size_t)KC;
  float* A      = ws + off; off += (size_t)KC * KC;
  size_t zbytes = (size_t)((ws + off) - zbase) * sizeof(float);
  hipMemsetAsync(zbase, 0, zbytes, stream);

  // 1. h = x @ W1  (WMMA f32 row-strip, W1 in LDS; x read exactly once)
  const int rtilesN = n >> 4;
  gemm_rs<IN_F, false><<<(rtilesN + 7) / 8, 256, 0, stream>>>(x, IN_F, W1, nullptr, Hh, n);

  // 2. degrees -> dinv
  deg_count<<<(E + 255) / 256, 256, 0, stream>>>(dstp, deg, E);
  deg_to_dinv<<<(n + 255) / 256, 256, 0, stream>>>(deg, n);

  // 3. normalized edge scatter-add (hot loop; h/agg are L2-resident at 12.8 MB)
  long long eth = (long long)E * 16;
  edge_agg<<<(unsigned)((eth + 255) / 256), 256, 0, stream>>>(srcp, dstp, deg, Hh, agg, E);

  // 4. x1 = relu(agg + dinv^2 h + b1), fused cluster-sum pooling
  long long nth = (long long)n * H_F;
  x1_pool<<<(unsigned)((nth + 255) / 256), 256, 0, stream>>>(agg, Hh, deg, b1, cid,
                                                             x1, sums, counts, n);
  xp_mean<<<(KC * H_F + 255) / 256, 256, 0, stream>>>(sums, counts, xp);

  // 5. pooled adjacency -> normalized transposed dense matrix
  a_build<<<(E + 255) / 256, 256, 0, stream>>>(srcp, dstp, cid, A, E);
  dinvp_k<<<(KC + 255) / 256, 256, 0, stream>>>(A, dinvp);
  an_build<<<(KC * KC + 255) / 256, 256, 0, stream>>>(A, dinvp, An);

  // 6. hp = xp @ W2 (W2 in LDS) ; xp2 = An @ hp + b2 (B from global/L2)
  const int rtilesK = KC >> 4;
  gemm_rs<H_F, false><<<(rtilesK + 7) / 8, 256, 0, stream>>>(xp, H_F, W2, nullptr, hp, KC);
  gemm_rs<KC, true><<<(rtilesK + 7) / 8, 256, 0, stream>>>(An, KC, hp, b2, xp2, KC);

  // 7. fused skip GEMM + cluster gather + alpha-scaled combine -> d_out
  skip_out<<<(rtilesN + 7) / 8, 256, 0, stream>>>(x1, Wsk, bsk, xp2, cid, alpha,
                                                  (float*)d_out, n);
}